// Shift_20426864460057
// MI455X (gfx1250) — compile-verified
//
#include <hip/hip_runtime.h>
#include <stdint.h>

// Channel-grouped spatial shift (3x3 positions, group size 42, remainder -> (0,0)).
// out[n,c,y,x] = in-bounds ? x[n,c,y+dy,x+dx] : 0, (dy,dx) uniform per channel.
//
// Pure data movement: ~403 MB HBM traffic, 0 FLOPs -> ~17us floor @ 23.3 TB/s.
// Strategy: stage each 16KB (n,c) plane into LDS with gfx1250 async b128
// global->LDS copies (all HBM reads aligned b128), resolve the +/-1 element
// x-shift via LDS word reads (free misalignment), write aligned b128 stores.

#define CHANNELS   384
#define GROUP_SIZE 42
#define H 64
#define W 64
#define PLANE (H * W)   // 4096 floats = 16 KB

__global__ __launch_bounds__(256)
void shift_async_lds_kernel(const float* __restrict__ in, float* __restrict__ out) {
    __shared__ float smem[PLANE];

    const int c = blockIdx.y;
    const int n = blockIdx.z;
    const size_t base = ((size_t)n * CHANNELS + (size_t)c) * PLANE;

    // Per-channel shift (uniform across the block -> pure SALU control flow).
    int g = c / GROUP_SIZE;
    if (g > 8) g = 0;                // remainder channels use kernel position (0,0)
    const int dy = g / 3 - 1;
    const int dx = g % 3 - 1;

    const int t = (int)threadIdx.x;

    // 32-bit DS-space address of smem: generic LDS pointer is
    // {SHARED_BASE aperture, 32-bit LDS offset}; addr[31:0] is the DS offset.
    // Deriving the asm operand from the real pointer also makes `smem` escape,
    // so the compiler cannot delete the LDS reads below.
    const unsigned smem_lds_base = (unsigned)(uintptr_t)(&smem[0]);

    // ---- Stage the full input plane into LDS via async b128 copies ----------
    // Each thread issues 4 lane-wise async loads: 256 thr * 4 * 16B = 16 KB.
    const char* gsrc = (const char*)(in + base);
    #pragma unroll
    for (int k = 0; k < 4; ++k) {
        unsigned byte_off = (unsigned)(t + 256 * k) * 16u;   // plane & LDS byte offset
        unsigned lds_addr = smem_lds_base + byte_off;
        unsigned long long gaddr = (unsigned long long)(gsrc + byte_off);
        asm volatile("global_load_async_to_lds_b128 %0, %1, off"
                     :: "v"(lds_addr), "v"(gaddr)
                     : "memory");
    }
    // Wait for this wave's async transfers to land in LDS, then sync the WG.
    asm volatile("s_wait_asynccnt 0x0" ::: "memory");
    __syncthreads();

    // ---- Shifted write-out: aligned float4 stores ---------------------------
    #pragma unroll
    for (int k = 0; k < 4; ++k) {
        const int v  = t + 256 * k;          // vec4 id within plane, 0..1023
        const int y  = v >> 4;               // output row
        const int x0 = (v & 15) * 4;         // output col of first element
        const int sy = y + dy;

        float4 r = make_float4(0.f, 0.f, 0.f, 0.f);
        if ((unsigned)sy < (unsigned)H) {
            const float* row = smem + sy * W;
            float tmp[4];
            #pragma unroll
            for (int j = 0; j < 4; ++j) {
                const int sx = x0 + j + dx;
                tmp[j] = ((unsigned)sx < (unsigned)W) ? row[sx] : 0.f;
            }
            r = make_float4(tmp[0], tmp[1], tmp[2], tmp[3]);
        }
        *(float4*)(out + base + (size_t)y * W + x0) = r;
    }
}

extern "C" void kernel_launch(void* const* d_in, const int* in_sizes, int n_in,
                              void* d_out, int out_size, void* d_ws, size_t ws_size,
                              hipStream_t stream) {
    (void)n_in; (void)out_size; (void)d_ws; (void)ws_size;
    const float* x = (const float*)d_in[0];
    float* out = (float*)d_out;

    const int N = in_sizes[0] / (CHANNELS * PLANE);   // batch (32 for the reference)

    dim3 grid(1, CHANNELS, N);
    dim3 block(256);
    hipLaunchKernelGGL(shift_async_lds_kernel, grid, block, 0, stream, x, out);
}